// RecurrentRGCN_71992241815987
// MI455X (gfx1250) — compile-verified
//
#include <hip/hip_runtime.h>
#include <math.h>

// ---------------------------------------------------------------------------
// RecurrentRGCN for MI455X (gfx1250, wave32).
// Key optimization: pull the neighbor GEMM out of the edge loop
// (segment_sum(x) @ W == segment_sum(x @ W)), turning 32 GFLOP/step of
// per-edge GEMM into a 4 GFLOP/step dense [N,200]@[200,200] GEMM done with
// V_WMMA_F32_16X16X4_F32 (exact fp32). Edge work becomes pure gather +
// f32 atomic scatter-add, which lives in the 192MB L2 (h is 40MB).
// A-strip staging uses gfx1250 GLOBAL_LOAD_ASYNC_TO_LDS_B128 (ASYNCcnt).
// ---------------------------------------------------------------------------

#define N_ENTS 50000
#define H_DIM  200
#define R2     500
#define T_STEPS 4
#define E_EDGES 400000
#define RRELU_SLOPE 0.22916666666666666f

typedef float v2f __attribute__((ext_vector_type(2)));
typedef float v8f __attribute__((ext_vector_type(8)));

__device__ __forceinline__ float sigmoidf(float x) {
    return 1.0f / (1.0f + __expf(-x));
}

__device__ __forceinline__ float wave_reduce_sum(float v) {
    #pragma unroll
    for (int off = 16; off > 0; off >>= 1)
        v += __shfl_xor(v, off, 32);
    return v;
}

// ---------------------------------------------------------------------------
// Row-wise L2 normalize: out[r] = in[r] / max(||in[r]||, 1e-12).  H_DIM cols.
// One wave32 per row, 8 rows per 256-thread block.
// ---------------------------------------------------------------------------
__global__ __launch_bounds__(256) void l2norm_rows_kernel(
    const float* __restrict__ in, float* __restrict__ out, int nrows) {
    int wave = threadIdx.x >> 5;
    int lane = threadIdx.x & 31;
    int row = blockIdx.x * 8 + wave;
    if (row >= nrows) return;
    const float* ir = in + (size_t)row * H_DIM;
    float* orow = out + (size_t)row * H_DIM;
    float s = 0.f;
    for (int j = lane; j < H_DIM; j += 32) { float v = ir[j]; s += v * v; }
    s = wave_reduce_sum(s);
    float inv = 1.0f / fmaxf(sqrtf(s), 1e-12f);
    for (int j = lane; j < H_DIM; j += 32) orow[j] = ir[j] * inv;
}

// ---------------------------------------------------------------------------
// Edge scatter: for each edge e, 50 threads each own a float4 column chunk.
//   preagg[dst] += h[src] + emb_rel[etype]     (RGCN pre-message, pre-GEMM)
//   relsum[r_seg] += h[r_to_e]                 (per-relation mean numerator)
//   deg[dst] += 1 ; relcnt[r_seg] += 1         (one thread per edge)
// ---------------------------------------------------------------------------
__global__ __launch_bounds__(256) void edge_scatter_kernel(
    const float* __restrict__ h, const float* __restrict__ emb_rel,
    const int* __restrict__ src, const int* __restrict__ dst,
    const int* __restrict__ etype, const int* __restrict__ rte,
    const int* __restrict__ rseg,
    float* __restrict__ preagg, float* __restrict__ deg,
    float* __restrict__ relsum, float* __restrict__ relcnt) {
    int tid = blockIdx.x * blockDim.x + threadIdx.x;
    const int CH = H_DIM / 4;  // 50 float4 chunks per row
    if (tid >= E_EDGES * CH) return;
    int e = tid / CH;
    int c = (tid - e * CH) * 4;

    int s = src[e], d = dst[e], t = etype[e];
    int rr = rte[e], sg = rseg[e];

    float4 hv = *(const float4*)(h + (size_t)s * H_DIM + c);
    float4 rv = *(const float4*)(emb_rel + (size_t)t * H_DIM + c);
    float* pd = preagg + (size_t)d * H_DIM + c;
    atomicAdd(pd + 0, hv.x + rv.x);
    atomicAdd(pd + 1, hv.y + rv.y);
    atomicAdd(pd + 2, hv.z + rv.z);
    atomicAdd(pd + 3, hv.w + rv.w);

    float4 mv = *(const float4*)(h + (size_t)rr * H_DIM + c);
    float* ps = relsum + (size_t)sg * H_DIM + c;
    atomicAdd(ps + 0, mv.x);
    atomicAdd(ps + 1, mv.y);
    atomicAdd(ps + 2, mv.z);
    atomicAdd(ps + 3, mv.w);

    if (c == 0) {
        atomicAdd(deg + d, 1.0f);
        atomicAdd(relcnt + sg, 1.0f);
    }
}

// ---------------------------------------------------------------------------
// x_cat[r] = [ emb_rel[r] , mean_of_segment(r) ]   ([R2, 2H])
// ---------------------------------------------------------------------------
__global__ __launch_bounds__(256) void xcat_kernel(
    const float* __restrict__ emb_rel, const float* __restrict__ relsum,
    const float* __restrict__ relcnt, float* __restrict__ xcat) {
    int tid = blockIdx.x * blockDim.x + threadIdx.x;
    if (tid >= R2 * H_DIM) return;
    int r = tid / H_DIM, j = tid - r * H_DIM;
    float cnt = relcnt[r];
    float mean = (cnt > 0.f) ? relsum[tid] / fmaxf(cnt, 1.0f) : 0.f;
    xcat[(size_t)r * (2 * H_DIM) + j] = emb_rel[tid];
    xcat[(size_t)r * (2 * H_DIM) + H_DIM + j] = mean;
}

// ---------------------------------------------------------------------------
// GRU cell over [R2, 2H] inputs + [R2, H] hidden (tiny: 0.36 GFLOP total).
// ---------------------------------------------------------------------------
__global__ __launch_bounds__(256) void gru_kernel(
    const float* __restrict__ xcat, const float* __restrict__ h0,
    const float* __restrict__ W_ih, const float* __restrict__ W_hh,
    const float* __restrict__ b_ih, const float* __restrict__ b_hh,
    float* __restrict__ out) {
    int tid = blockIdx.x * blockDim.x + threadIdx.x;
    if (tid >= R2 * H_DIM) return;
    int r = tid / H_DIM, j = tid - r * H_DIM;
    const int K2 = 2 * H_DIM;

    float ir = b_ih[j], iz = b_ih[j + H_DIM], inn = b_ih[j + 2 * H_DIM];
    const float* xr = xcat + (size_t)r * K2;
    const float* wr = W_ih + (size_t)j * K2;
    const float* wz = W_ih + (size_t)(j + H_DIM) * K2;
    const float* wn = W_ih + (size_t)(j + 2 * H_DIM) * K2;
    for (int k = 0; k < K2; ++k) {
        float x = xr[k];
        ir = fmaf(x, wr[k], ir); iz = fmaf(x, wz[k], iz); inn = fmaf(x, wn[k], inn);
    }
    float hr = b_hh[j], hz = b_hh[j + H_DIM], hn = b_hh[j + 2 * H_DIM];
    const float* hrow = h0 + (size_t)r * H_DIM;
    const float* vr = W_hh + (size_t)j * H_DIM;
    const float* vz = W_hh + (size_t)(j + H_DIM) * H_DIM;
    const float* vn = W_hh + (size_t)(j + 2 * H_DIM) * H_DIM;
    for (int k = 0; k < H_DIM; ++k) {
        float x = hrow[k];
        hr = fmaf(x, vr[k], hr); hz = fmaf(x, vz[k], hz); hn = fmaf(x, vn[k], hn);
    }
    float rg = sigmoidf(ir + hr);
    float zg = sigmoidf(iz + hz);
    float ng = tanhf(inn + rg * hn);
    out[tid] = (1.0f - zg) * ng + zg * hrow[j];
}

// ---------------------------------------------------------------------------
// fp32 WMMA GEMM: C[M x 200] = A[M x 200] @ B[200 x 200], all row-major.
// Block = 128 threads (4 wave32s), each wave owns a 16x16 output tile.
// grid = (ceil(M/64), 13 col tiles).
//
// A strip (64x200, pad-204 stride) staged via GLOBAL_LOAD_ASYNC_TO_LDS_B128:
// per-lane 16B global->LDS copies that bypass VGPRs (ASYNCcnt tracked),
// overlapping with the VALU gather of the B tile.  Out-of-range rows of the
// last M-block read adjacent workspace (valid memory); their output tiles
// are skipped by the wave-uniform mbase<M guard, so garbage never escapes.
// The low 32 bits of a generic pointer to __shared__ are the LDS byte
// offset (flat->LDS aperture translation truncates to addr[31:0]), giving
// the async instruction its VDST address; all chunks are 16B aligned
// (816 = 51*16).
//
// B column tile (200x16) stored as K-adjacent float2 pairs so each B
// fragment is one ds_load_b64.  LDS total ~65 KB -> 4 blocks/WGP.
// K-loop fully unrolled: 50x V_WMMA_F32_16X16X4_F32 with immediate-offset
// DS loads.  Fragment layouts per CDNA5 ISA 7.12.2.
// ---------------------------------------------------------------------------
#define APAD 204
__global__ __launch_bounds__(128) void wmma_gemm200_kernel(
    const float* __restrict__ A, const float* __restrict__ B,
    float* __restrict__ C, int M) {
    __shared__ float As[64 * APAD];
    __shared__ v2f   Bs2[(H_DIM / 2) * 16];

    int tid = threadIdx.x;
    int m0 = blockIdx.x * 64;
    int n0 = blockIdx.y * 16;

    // async stage of A strip: 64 rows x 50 float4 chunks, 25 per thread
    const int CH4 = H_DIM / 4;  // 50
    for (int idx = tid; idx < 64 * CH4; idx += 128) {
        int r = idx / CH4;
        int k4 = idx - r * CH4;
        unsigned lds_byte = (unsigned)(size_t)(&As[r * APAD + k4 * 4]);
        const float* gsrc = A + (size_t)(m0 + r) * H_DIM + k4 * 4;
        asm volatile("global_load_async_to_lds_b128 %0, %1, off"
                     :: "v"(lds_byte), "v"(gsrc) : "memory");
    }
    // cooperative stage of B column tile as {B[2p][c], B[2p+1][c]} pairs
    for (int idx = tid; idx < (H_DIM / 2) * 16; idx += 128) {
        int p = idx >> 4, c = idx & 15;
        int gc = n0 + c;
        v2f val;
        val.x = (gc < H_DIM) ? B[(size_t)(2 * p) * H_DIM + gc] : 0.f;
        val.y = (gc < H_DIM) ? B[(size_t)(2 * p + 1) * H_DIM + gc] : 0.f;
        Bs2[idx] = val;
    }
    asm volatile("s_wait_asynccnt 0" ::: "memory");  // own async copies done
    __syncthreads();                                 // all waves' copies done

    int wave = tid >> 5;
    int lane = tid & 31;
    int l16 = lane & 15;
    int hi = (lane >= 16) ? 2 : 0;   // K offset for the high half-wave
    int rb = wave * 16;              // wave's row base inside the strip

    const float* arow = &As[(rb + l16) * APAD + hi];  // lane-fixed A base
    const v2f*   brow = &Bs2[(hi >> 1) * 16 + l16];   // lane-fixed B base

    v8f acc = {};
    #pragma unroll
    for (int kk = 0; kk < H_DIM; kk += 4) {
        v2f a = *(const v2f*)(arow + kk);             // ds_load_b64, imm offset
        v2f b = brow[(kk >> 1) * 16];                 // ds_load_b64, imm offset
        acc = __builtin_amdgcn_wmma_f32_16x16x4_f32(
            false, a, false, b, (short)0, acc, false, false);
    }

    int rowoff = (lane >= 16) ? 8 : 0;
    int col = n0 + l16;
    int mbase = m0 + rb + rowoff;
    if (mbase < M) {                 // wave-uniform (M % 16 == 0)
        if (col < H_DIM) {           // divergent only in last col tile
            float* cp = C + (size_t)mbase * H_DIM + col;
            #pragma unroll
            for (int r = 0; r < 8; ++r)
                cp[(size_t)r * H_DIM] = acc[r];   // 8 stores, imm offsets
        }
    }
}

// ---------------------------------------------------------------------------
// Fused epilogue per entity row (one wave32 per row):
//   cur  = l2norm( rrelu( g1 * (deg>0 ? 1/max(deg,1) : 0) ) )
//   tw   = sigmoid(g2 + time_gate_bias)
//   hnew = l2norm( tw*cur + (1-tw)*h )  -> evolve[t] and next-step h
// ---------------------------------------------------------------------------
__global__ __launch_bounds__(256) void finalize_kernel(
    const float* __restrict__ g1, const float* __restrict__ g2,
    const float* __restrict__ deg, const float* __restrict__ tgb,
    const float* __restrict__ h, float* __restrict__ evolve,
    float* __restrict__ hnext) {
    int wave = threadIdx.x >> 5;
    int lane = threadIdx.x & 31;
    int row = blockIdx.x * 8 + wave;
    if (row >= N_ENTS) return;
    size_t base = (size_t)row * H_DIM;

    float dg = deg[row];
    float nrm = (dg > 0.f) ? 1.0f / fmaxf(dg, 1.0f) : 0.f;

    float cv[7];
    float s1 = 0.f;
    #pragma unroll
    for (int i = 0; i < 7; ++i) {
        int j = lane + i * 32;
        float c = 0.f;
        if (j < H_DIM) {
            c = g1[base + j] * nrm;
            c = (c >= 0.f) ? c : RRELU_SLOPE * c;
        }
        cv[i] = c;
        s1 += c * c;
    }
    s1 = wave_reduce_sum(s1);
    float inv1 = 1.0f / fmaxf(sqrtf(s1), 1e-12f);

    float vv[7];
    float s2 = 0.f;
    #pragma unroll
    for (int i = 0; i < 7; ++i) {
        int j = lane + i * 32;
        float v = 0.f;
        if (j < H_DIM) {
            float tw = sigmoidf(g2[base + j] + tgb[j]);
            v = tw * (cv[i] * inv1) + (1.0f - tw) * h[base + j];
        }
        vv[i] = v;
        s2 += v * v;
    }
    s2 = wave_reduce_sum(s2);
    float inv2 = 1.0f / fmaxf(sqrtf(s2), 1e-12f);

    #pragma unroll
    for (int i = 0; i < 7; ++i) {
        int j = lane + i * 32;
        if (j < H_DIM) {
            float o = vv[i] * inv2;
            evolve[base + j] = o;
            hnext[base + j] = o;
        }
    }
}

// ---------------------------------------------------------------------------
extern "C" void kernel_launch(void* const* d_in, const int* in_sizes, int n_in,
                              void* d_out, int out_size, void* d_ws, size_t ws_size,
                              hipStream_t stream) {
    const float* dyn     = (const float*)d_in[0];
    const float* emb_rel = (const float*)d_in[1];
    const float* W_ih    = (const float*)d_in[2];
    const float* W_hh    = (const float*)d_in[3];
    const float* b_ih    = (const float*)d_in[4];
    const float* b_hh    = (const float*)d_in[5];
    const float* w_nb    = (const float*)d_in[6];
    const float* tgw     = (const float*)d_in[7];
    const float* tgb     = (const float*)d_in[8];
    const int*   src     = (const int*)d_in[9];
    const int*   dst     = (const int*)d_in[10];
    const int*   etype   = (const int*)d_in[11];
    const int*   rte     = (const int*)d_in[12];
    const int*   rseg    = (const int*)d_in[13];

    float* out = (float*)d_out;                  // evolve [T,N,H] then h0 [R2,H]
    float* evolve = out;
    float* h0_out = out + (size_t)T_STEPS * N_ENTS * H_DIM;

    // ---- workspace carve-up (floats) ----
    float* ws = (float*)d_ws;
    const size_t NH = (size_t)N_ENTS * H_DIM;    // 10M
    const size_t RH = (size_t)R2 * H_DIM;        // 100k
    float* hA      = ws;            ws += NH;
    float* hB      = ws;            ws += NH;
    float* preagg  = ws;            ws += NH;    // |-- contiguous zeroed region
    float* deg     = ws;            ws += N_ENTS;
    float* relsum  = ws;            ws += RH;
    float* relcnt  = ws;            ws += 512;   // --|
    float* g1      = ws;            ws += NH;
    float* g2      = ws;            ws += NH;
    float* xcat    = ws;            ws += (size_t)R2 * 2 * H_DIM;
    float* h0A     = ws;            ws += RH;
    float* h0B     = ws;            ws += RH;
    float* gru_tmp = ws;            ws += RH;
    const size_t zero_bytes = (NH + N_ENTS + RH + 512) * sizeof(float);

    // ---- init ----
    l2norm_rows_kernel<<<(N_ENTS + 7) / 8, 256, 0, stream>>>(dyn, hA, N_ENTS);
    (void)hipMemcpyAsync(h0A, emb_rel, RH * sizeof(float),
                         hipMemcpyDeviceToDevice, stream);

    float* h_cur = hA;  float* h_nxt = hB;
    float* h0_cur = h0A; float* h0_nxt = h0B;

    const int CH = H_DIM / 4;
    const int edge_blocks = (E_EDGES * CH + 255) / 256;
    const dim3 gemm_grid((N_ENTS + 63) / 64, (H_DIM + 15) / 16);

    for (int t = 0; t < T_STEPS; ++t) {
        (void)hipMemsetAsync(preagg, 0, zero_bytes, stream);

        edge_scatter_kernel<<<edge_blocks, 256, 0, stream>>>(
            h_cur, emb_rel,
            src + (size_t)t * E_EDGES, dst + (size_t)t * E_EDGES,
            etype + (size_t)t * E_EDGES, rte + (size_t)t * E_EDGES,
            rseg + (size_t)t * E_EDGES,
            preagg, deg, relsum, relcnt);

        // GRU branch (independent of the entity branch)
        xcat_kernel<<<(R2 * H_DIM + 255) / 256, 256, 0, stream>>>(
            emb_rel, relsum, relcnt, xcat);
        gru_kernel<<<(R2 * H_DIM + 255) / 256, 256, 0, stream>>>(
            xcat, h0_cur, W_ih, W_hh, b_ih, b_hh, gru_tmp);
        l2norm_rows_kernel<<<(R2 + 7) / 8, 256, 0, stream>>>(gru_tmp, h0_nxt, R2);

        // entity branch: two fp32 WMMA GEMMs + fused gating epilogue
        wmma_gemm200_kernel<<<gemm_grid, 128, 0, stream>>>(preagg, w_nb, g1, N_ENTS);
        wmma_gemm200_kernel<<<gemm_grid, 128, 0, stream>>>(h_cur, tgw, g2, N_ENTS);
        finalize_kernel<<<(N_ENTS + 7) / 8, 256, 0, stream>>>(
            g1, g2, deg, tgb, h_cur,
            evolve + (size_t)t * NH, h_nxt);

        { float* tmp = h_cur; h_cur = h_nxt; h_nxt = tmp; }
        { float* tmp = h0_cur; h0_cur = h0_nxt; h0_nxt = tmp; }
    }

    (void)hipMemcpyAsync(h0_out, h0_cur, RH * sizeof(float),
                         hipMemcpyDeviceToDevice, stream);
}